// RelationalAttentionLayer_88055419502884
// MI455X (gfx1250) — compile-verified
//
#include <hip/hip_runtime.h>
#include <hip/hip_bf16.h>
#include <math.h>

typedef __attribute__((ext_vector_type(2))) float v2f;
typedef __attribute__((ext_vector_type(8))) float v8f;

__device__ __forceinline__ float atomAddF(float* p, float v) {
  return __hip_atomic_fetch_add(p, v, __ATOMIC_RELAXED, __HIP_MEMORY_SCOPE_AGENT);
}

// ---------------------------------------------------------------------------
// fp32 WMMA GEMM:  C[m,n] = sum_k A[m,k]*B[k,n] + bias[n]
// One wave computes a 16x64 tile: 4 x V_WMMA_F32_16X16X4_F32 per K-step,
// sharing one A fragment (4x operand reuse).
// A-frag (ISA 16x4 f32): lanes 0-15 rows, VGPR0/1 = K{0,1}; lanes 16-31 =
// K{2,3} -> one aligned float2 load per lane. B-frag mirrors (4x16).
// C/D: VGPR r -> row m0+r (lanes 0-15) / m0+8+r (lanes 16-31), col = lane&15.
// Fast path (fully in-bounds tile, wave-uniform test -> EXEC all-1s at every
// WMMA) has zero masking VALU; ragged tiles take a masked slow path.
// dup_off>0 duplicates stores at column offset dup_off (head broadcast).
// ---------------------------------------------------------------------------
__global__ void wmma_gemm_f32(const float* __restrict__ A, int lda,
                              const float* __restrict__ B, int ldb,
                              const float* __restrict__ bias,
                              float* __restrict__ C, int ldc, int dup_off,
                              int M, int N, int K)
{
  const int nt64   = (N + 63) >> 6;
  const int mtiles = (M + 15) >> 4;
  const int wid  = (int)((blockIdx.x * blockDim.x + threadIdx.x) >> 5);
  if (wid >= nt64 * mtiles) return;              // uniform per-wave exit
  const int lane = (int)(threadIdx.x & 31);
  const int nt = wid % nt64;
  const int mt = wid / nt64;
  const int m0 = mt << 4;
  const int n0 = nt << 6;
  const int lh = lane & 15;
  const int kh = (lane >> 4) << 1;               // 0 (lanes 0-15) or 2 (16-31)
  const int am = m0 + lh;

  v8f acc[4];
#pragma unroll
  for (int s = 0; s < 4; ++s) acc[s] = (v8f){0.f,0.f,0.f,0.f,0.f,0.f,0.f,0.f};

  if ((m0 + 16 <= M) && (n0 + 64 <= N)) {
    // ---------------- fast path: no masking in the hot loop ----------------
    const float* Ap = A + (size_t)am * lda + kh;         // 8B-aligned float2
    const float* Bp = B + (size_t)kh * ldb + n0 + lh;
#pragma unroll 2
    for (int k = 0; k < K; k += 4) {
      v2f a = *(const v2f*)(Ap + k);
      const float* b0 = Bp + (size_t)k * ldb;            // row kh+k
      const float* b1 = b0 + ldb;                        // row kh+k+1
      v2f bb0 = { b0[0],  b1[0]  };
      v2f bb1 = { b0[16], b1[16] };
      v2f bb2 = { b0[32], b1[32] };
      v2f bb3 = { b0[48], b1[48] };
      acc[0] = __builtin_amdgcn_wmma_f32_16x16x4_f32(false, a, false, bb0,
                                                     (short)0, acc[0], false, false);
      acc[1] = __builtin_amdgcn_wmma_f32_16x16x4_f32(false, a, false, bb1,
                                                     (short)0, acc[1], false, false);
      acc[2] = __builtin_amdgcn_wmma_f32_16x16x4_f32(false, a, false, bb2,
                                                     (short)0, acc[2], false, false);
      acc[3] = __builtin_amdgcn_wmma_f32_16x16x4_f32(false, a, false, bb3,
                                                     (short)0, acc[3], false, false);
    }
  } else {
    // ---------------- masked path for ragged edge tiles --------------------
    const float amask = (am < M) ? 1.0f : 0.0f;
    const int   amc   = (am < M) ? am : (M - 1);
    const float* Ap = A + (size_t)amc * lda + kh;
    float bmask[4]; int bnc[4];
#pragma unroll
    for (int s = 0; s < 4; ++s) {
      int bn = n0 + (s << 4) + lh;
      bmask[s] = (bn < N) ? 1.0f : 0.0f;
      bnc[s]   = (bn < N) ? bn : (N - 1);
    }
    for (int k = 0; k < K; k += 4) {
      v2f a;
      a.x = Ap[k]     * amask;
      a.y = Ap[k + 1] * amask;
      const float* b0 = B + (size_t)(kh + k) * ldb;
      const float* b1 = b0 + ldb;
#pragma unroll
      for (int s = 0; s < 4; ++s) {
        v2f bb;
        bb.x = b0[bnc[s]] * bmask[s];
        bb.y = b1[bnc[s]] * bmask[s];
        acc[s] = __builtin_amdgcn_wmma_f32_16x16x4_f32(false, a, false, bb,
                                                       (short)0, acc[s], false, false);
      }
    }
  }

  // ---------------- store (divergence is fine after the WMMAs) -------------
  const int mbase = m0 + ((lane >> 4) << 3);
#pragma unroll
  for (int s = 0; s < 4; ++s) {
    int bn = n0 + (s << 4) + lh;
    if (bn < N) {
      float bv = bias ? bias[bn] : 0.0f;
#pragma unroll
      for (int r = 0; r < 8; ++r) {
        int m = mbase + r;
        if (m < M) {
          float v = acc[s][r] + bv;
          float* p = C + (size_t)m * ldc + bn;
          p[0] = v;
          if (dup_off) p[dup_off] = v;
        }
      }
    }
  }
}

// ---------------------------------------------------------------------------
// Per-node attention partial dots. HsHd is [N,400] (cols 0-199 = Hs,
// 200-399 = Hd). Produces AsAd[n*4+{0,1,2,3}] = {att.Hs_h0, att.Hs_h1,
// att.Hd_h0, att.Hd_h1}. One wave per node.
// ---------------------------------------------------------------------------
__global__ void node_att_kernel(const float* __restrict__ HsHd,
                                const float* __restrict__ attw,  // [200]
                                float* __restrict__ AsAd, int Nn)
{
  const int wid  = (int)((blockIdx.x * blockDim.x + threadIdx.x) >> 5);
  const int lane = (int)(threadIdx.x & 31);
  if (wid >= Nn) return;
  const float* p = HsHd + (size_t)wid * 400;
  float a0 = 0.f, a1 = 0.f, a2 = 0.f, a3 = 0.f;
#pragma unroll
  for (int it = 0; it < 13; ++it) {
    int j = lane + it * 32;
    if (j < 400) {
      int aj = (j < 200) ? j : (j - 200);
      float v = attw[aj] * p[j];
      if      (j < 100) a0 += v;
      else if (j < 200) a1 += v;
      else if (j < 300) a2 += v;
      else              a3 += v;
    }
  }
  for (int off = 16; off > 0; off >>= 1) {
    a0 += __shfl_xor(a0, off, 32);
    a1 += __shfl_xor(a1, off, 32);
    a2 += __shfl_xor(a2, off, 32);
    a3 += __shfl_xor(a3, off, 32);
  }
  if (lane == 0) {
    float* q = AsAd + (size_t)wid * 4;
    q[0] = a0; q[1] = a1; q[2] = a2; q[3] = a3;
  }
}

// Per-relation attention partial dots: Ag[t*2+h] = att_h . Gp[t, h*100:...]
__global__ void rel_att_kernel(const float* __restrict__ Gp,
                               const float* __restrict__ attw,
                               float* __restrict__ Ag, int R)
{
  const int wid  = (int)((blockIdx.x * blockDim.x + threadIdx.x) >> 5);
  const int lane = (int)(threadIdx.x & 31);
  if (wid >= R) return;
  const float* p = Gp + (size_t)wid * 200;
  float a0 = 0.f, a1 = 0.f;
  for (int j = lane; j < 200; j += 32) {
    float v = attw[j] * p[j];
    if (j < 100) a0 += v; else a1 += v;
  }
  for (int off = 16; off > 0; off >>= 1) {
    a0 += __shfl_xor(a0, off, 32);
    a1 += __shfl_xor(a1, off, 32);
  }
  if (lane == 0) { Ag[(size_t)wid * 2] = a0; Ag[(size_t)wid * 2 + 1] = a1; }
}

// Edge pass 1: logits -> leakyrelu -> exp; accumulate segment sums over row.
__global__ void edge_logit_kernel(const int* __restrict__ row,
                                  const int* __restrict__ col,
                                  const int* __restrict__ etype,
                                  const float* __restrict__ AsAd,
                                  const float* __restrict__ Ag,
                                  float* __restrict__ ew,
                                  float* __restrict__ rsum, int E)
{
  const int e = (int)(blockIdx.x * blockDim.x + threadIdx.x);
  if (e >= E) return;
  const int r = row[e], c = col[e], t = etype[e];
  v2f as = *(const v2f*)(AsAd + (size_t)r * 4);       // {As_h0, As_h1}
  v2f ad = *(const v2f*)(AsAd + (size_t)c * 4 + 2);   // {Ad_h0, Ad_h1}
  v2f ag = *(const v2f*)(Ag   + (size_t)t * 2);       // {Ag_h0, Ag_h1}
  v2f w;
#pragma unroll
  for (int h = 0; h < 2; ++h) {
    float a = as[h] + ad[h] + ag[h];
    a = (a >= 0.f) ? a : 0.01f * a;                   // LeakyReLU(0.01)
    w[h] = expf(a);
    atomAddF(&rsum[(size_t)r * 2 + h], w[h]);
  }
  *(v2f*)(ew + (size_t)e * 2) = w;
}

// Edge pass 2: alpha = ew/rsum[row]; scatter alpha*c_ijk into out[col].
// One wave per edge; lanes stride the 200 feature columns.
__global__ void edge_agg_kernel(const int* __restrict__ row,
                                const int* __restrict__ col,
                                const int* __restrict__ etype,
                                const float* __restrict__ HsHd,
                                const float* __restrict__ Gp,
                                const float* __restrict__ ew,
                                const float* __restrict__ rsum,
                                float* __restrict__ out, int E)
{
  const int wid  = (int)((blockIdx.x * blockDim.x + threadIdx.x) >> 5);
  const int lane = (int)(threadIdx.x & 31);
  if (wid >= E) return;
  const int r = row[wid], c = col[wid], t = etype[wid];
  v2f wv = *(const v2f*)(ew   + (size_t)wid * 2);
  v2f rs = *(const v2f*)(rsum + (size_t)r * 2);
  const float al0 = wv.x / rs.x;
  const float al1 = wv.y / rs.y;
  const float* ps = HsHd + (size_t)r * 400;        // Hs[row]
  const float* pd = HsHd + (size_t)c * 400 + 200;  // Hd[col]
  const float* pg = Gp   + (size_t)t * 200;        // Gp[type]
  float* po = out + (size_t)c * 200;
#pragma unroll
  for (int it = 0; it < 7; ++it) {
    int j = lane + it * 32;
    if (j < 200) {
      float cv = ps[j] + pd[j] + pg[j];
      float alpha = (j < 100) ? al0 : al1;
      atomAddF(&po[j], alpha * cv);
    }
  }
}

__global__ void tanh_kernel(float* __restrict__ out, long n)
{
  long i = (long)blockIdx.x * blockDim.x + threadIdx.x;
  if (i < n) out[i] = tanhf(out[i]);
}

// ---------------------------------------------------------------------------
extern "C" void kernel_launch(void* const* d_in, const int* in_sizes, int n_in,
                              void* d_out, int out_size, void* d_ws, size_t ws_size,
                              hipStream_t stream)
{
  const float* x     = (const float*)d_in[0];
  const float* h     = (const float*)d_in[1];
  const float* g     = (const float*)d_in[2];
  const int*   eidx  = (const int*)  d_in[3];
  const int*   etype = (const int*)  d_in[4];
  const float* fc1_w = (const float*)d_in[5];
  const float* fc1_b = (const float*)d_in[6];
  const float* att_w = (const float*)d_in[7];
  const float* rel_w = (const float*)d_in[8];
  const float* rel_b = (const float*)d_in[9];
  const float* ent_w = (const float*)d_in[10];
  const float* ent_b = (const float*)d_in[11];

  const int INIT = 100, INH = 200, ING = 200, OUT = 100, HO = 200;
  const int N = in_sizes[0] / INIT;
  const int R = in_sizes[2] / ING;
  const int E = in_sizes[4];

  const int* row = eidx;        // edge_idx[0]
  const int* col = eidx + E;    // edge_idx[1]

  // workspace layout (~88 MB)
  float* HsHd = (float*)d_ws;                    // N*400
  float* Gp   = HsHd + (size_t)N * 400;          // R*200
  float* AsAd = Gp   + (size_t)R * 200;          // N*4
  float* Ag   = AsAd + (size_t)N * 4;            // R*2
  float* ew   = Ag   + (size_t)R * 2;            // E*2
  float* rsum = ew   + (size_t)E * 2;            // N*2

  float* hprime = (float*)d_out;                 // N*200
  float* gprime = hprime + (size_t)N * HO;       // R*100

  hipMemsetAsync(rsum, 0, (size_t)N * 2 * sizeof(float), stream);

  // --- dense GEMMs on the WMMA path -------------------------------------
  struct GemmCfg { const float* A; int lda; const float* B; int ldb;
                   const float* bias; float* C; int ldc; int dup;
                   int M, Nc, K; };
  const GemmCfg gemms[5] = {
    { h, INH, fc1_w,                        HO, nullptr, HsHd,       400, 0,   N, HO,  INH }, // Hs
    { h, INH, fc1_w + (size_t)INH * HO,     HO, nullptr, HsHd + 200, 400, 0,   N, HO,  INH }, // Hd
    { g, ING, fc1_w + (size_t)2 * INH * HO, HO, fc1_b,   Gp,         HO,  0,   R, HO,  ING }, // Gp
    { x, INIT, ent_w,                       OUT, ent_b,  hprime,     HO,  OUT, N, OUT, INIT}, // x@ent_w (dup heads)
    { g, ING, rel_w,                        OUT, rel_b,  gprime,     OUT, 0,   R, OUT, ING }, // g'
  };
  for (int i = 0; i < 5; ++i) {
    const GemmCfg& cf = gemms[i];
    long waves = (long)((cf.Nc + 63) / 64) * ((cf.M + 15) / 16);
    unsigned grid = (unsigned)((waves + 3) / 4);           // 4 waves / block
    wmma_gemm_f32<<<dim3(grid), dim3(128), 0, stream>>>(
        cf.A, cf.lda, cf.B, cf.ldb, cf.bias, cf.C, cf.ldc, cf.dup,
        cf.M, cf.Nc, cf.K);
  }

  // --- per-node / per-relation attention partials -----------------------
  {
    unsigned grid = (unsigned)((N + 7) / 8);               // 8 waves / block
    node_att_kernel<<<dim3(grid), dim3(256), 0, stream>>>(HsHd, att_w, AsAd, N);
  }
  {
    unsigned grid = (unsigned)(((long)R * 32 + 255) / 256);
    rel_att_kernel<<<dim3(grid), dim3(256), 0, stream>>>(Gp, att_w, Ag, R);
  }

  // --- edge passes ------------------------------------------------------
  {
    unsigned grid = (unsigned)(((long)E + 255) / 256);
    edge_logit_kernel<<<dim3(grid), dim3(256), 0, stream>>>(
        row, col, etype, AsAd, Ag, ew, rsum, E);
  }
  {
    unsigned grid = (unsigned)(((long)E + 7) / 8);         // wave per edge
    edge_agg_kernel<<<dim3(grid), dim3(256), 0, stream>>>(
        row, col, etype, HsHd, Gp, ew, rsum, hprime, E);
  }

  // --- final activation -------------------------------------------------
  {
    long n = (long)N * HO;
    unsigned grid = (unsigned)((n + 255) / 256);
    tanh_kernel<<<dim3(grid), dim3(256), 0, stream>>>(hprime, n);
  }
}